// Graph_test_36206574305989
// MI455X (gfx1250) — compile-verified
//
#include <hip/hip_runtime.h>

typedef float v2f __attribute__((ext_vector_type(2)));
typedef float v8f __attribute__((ext_vector_type(8)));

#define LREL(v) ((v) > 0.0f ? (v) : 0.01f * (v))
#define NNODE 78
#define MAXE 4096
#define TSTR 388   // padded LDS row stride (floats): 388 % 64 == 4 -> no bank conflicts

// ---------------------------------------------------------------------------
// CSR build: sort edges by target (stable -> deterministic reduction order).
// E ~ 1756, so a single thread is fine and fully deterministic.
// ---------------------------------------------------------------------------
__global__ void build_csr_kernel(const int* __restrict__ ei, int E,
                                 int* __restrict__ row_ptr,
                                 int* __restrict__ colbuf) {
    if (blockIdx.x != 0 || threadIdx.x != 0) return;
    int cnt[NNODE];
    for (int n = 0; n < NNODE; ++n) cnt[n] = 0;
    for (int e = 0; e < E; ++e) cnt[ei[E + e]]++;
    int run = 0;
    for (int n = 0; n < NNODE; ++n) {
        row_ptr[n] = run;
        run += cnt[n];
        cnt[n] = row_ptr[n];
    }
    row_ptr[NNODE] = run;
    for (int e = 0; e < E; ++e) {
        int s = ei[e], t = ei[E + e];
        colbuf[cnt[t]++] = s;
    }
}

// ---------------------------------------------------------------------------
// Encoder: per 16-row tile: LayerNorm(384) in LDS, then fp32 WMMA GEMM
// (V_WMMA_F32_16X16X4_F32, K stepped by 4) against enc_w^T (staged in LDS,
// zero-padded row for N>=20), + bias, leaky_relu -> x[:,58:78];
// also copies cli_data into x[:,0:58].
// Block = 64 threads (2 waves): wave0 -> N cols 0..15, wave1 -> cols 16..19.
// ---------------------------------------------------------------------------
__global__ __launch_bounds__(64)
void encode_kernel(const float* __restrict__ radio, const float* __restrict__ cli,
                   const float* __restrict__ ln1g, const float* __restrict__ ln1b,
                   const float* __restrict__ encw, const float* __restrict__ encb,
                   float* __restrict__ xout) {
    __shared__ float tile[16 * TSTR];   // A: 16 normalized rows of radio
    __shared__ float bmat[21 * TSTR];   // B: enc_w rows 0..19, row 20 = zeros (pad)
    __shared__ float part[64];
    __shared__ float mean_s[16];
    __shared__ float rstd_s[16];

    const int tid = threadIdx.x;
    const int m0  = blockIdx.x * 16;

    // Stage A (radio rows) and B (enc_w + zero pad row) into LDS.
    for (int idx = tid; idx < 16 * 384; idx += 64) {
        int r = idx / 384, c = idx - r * 384;
        tile[r * TSTR + c] = radio[(size_t)(m0 + r) * 384 + c];
    }
    for (int idx = tid; idx < 20 * 384; idx += 64) {
        int r = idx / 384, c = idx - r * 384;
        bmat[r * TSTR + c] = encw[idx];
    }
    for (int c = tid; c < 384; c += 64) bmat[20 * TSTR + c] = 0.0f;
    __syncthreads();

    // Row mean (4 partial threads per row).
    {
        int r = tid & 15, seg = tid >> 4;
        float s = 0.f;
        for (int c = seg; c < 384; c += 4) s += tile[r * TSTR + c];
        part[tid] = s;
    }
    __syncthreads();
    if (tid < 16)
        mean_s[tid] = (part[tid] + part[16 + tid] + part[32 + tid] + part[48 + tid]) * (1.0f / 384.0f);
    __syncthreads();

    // Row variance (biased, like jnp.var).
    {
        int r = tid & 15, seg = tid >> 4;
        float m = mean_s[r], s = 0.f;
        for (int c = seg; c < 384; c += 4) { float d = tile[r * TSTR + c] - m; s += d * d; }
        part[tid] = s;
    }
    __syncthreads();
    if (tid < 16) {
        float v = (part[tid] + part[16 + tid] + part[32 + tid] + part[48 + tid]) * (1.0f / 384.0f);
        rstd_s[tid] = rsqrtf(v + 1e-5f);
    }
    __syncthreads();

    // Normalize in place (apply gamma/beta).
    for (int idx = tid; idx < 16 * 384; idx += 64) {
        int r = idx / 384, c = idx - r * 384;
        tile[r * TSTR + c] = (tile[r * TSTR + c] - mean_s[r]) * rstd_s[r] * ln1g[c] + ln1b[c];
    }
    __syncthreads();

    // fp32 WMMA: D(16x16) += A(16x4) x B(4x16), K = 0..383 step 4.
    // A layout: lanes 0-15 hold {K0,K1} of row M=lane; lanes 16-31 hold {K2,K3}.
    // B mirrors A with N across lanes.
    const int wave = tid >> 5;
    const int lane = tid & 31;
    const int l15  = lane & 15;
    const int koff = (lane < 16) ? 0 : 2;
    const int ncol = wave * 16 + l15;
    const int brow = (ncol < 20) ? ncol : 20;   // padded zero row for N >= 20

    const int abase = l15 * TSTR + koff;
    const int bbase = brow * TSTR + koff;

    v8f acc = {};
    for (int k0 = 0; k0 < 384; k0 += 4) {
        v2f a = *(const v2f*)&tile[abase + k0];   // ds_load_b64, conflict-free
        v2f b = *(const v2f*)&bmat[bbase + k0];   // ds_load_b64, conflict-free
        acc = __builtin_amdgcn_wmma_f32_16x16x4_f32(
                  false, a, false, b, (short)0, acc, false, false);
    }

    // Epilogue: C/D layout: VGPR j -> row j (lanes 0-15) / row j+8 (lanes 16-31), N = lane%16.
    if (ncol < 20) {
        float bb = encb[ncol];
        for (int j = 0; j < 8; ++j) {
            int row = j + ((lane >= 16) ? 8 : 0);
            float v = acc[j] + bb;
            v = LREL(v);
            xout[(size_t)(m0 + row) * NNODE + 58 + ncol] = v;
        }
    }

    // Copy cli_data into x[:, 0:58].
    for (int idx = tid; idx < 16 * 58; idx += 64) {
        int r = idx / 58, c = idx - r * 58;
        xout[(size_t)(m0 + r) * NNODE + c] = cli[(size_t)(m0 + r) * 58 + c];
    }
}

// ---------------------------------------------------------------------------
// Fused per-graph kernel: TransformerConv x2 + LayerNorm(78) + Linear(78->2).
// One 128-thread block per graph; everything stays in LDS.
// ---------------------------------------------------------------------------
__global__ __launch_bounds__(128)
void graph_kernel(const float* __restrict__ x,
                  const int* __restrict__ row_ptr, const int* __restrict__ colg, int E,
                  const float* __restrict__ q1w, const float* __restrict__ q1b,
                  const float* __restrict__ k1w, const float* __restrict__ k1b,
                  const float* __restrict__ v1w, const float* __restrict__ v1b,
                  const float* __restrict__ s1w, const float* __restrict__ s1b,
                  const float* __restrict__ q2w, const float* __restrict__ q2b,
                  const float* __restrict__ k2w, const float* __restrict__ k2b,
                  const float* __restrict__ v2w, const float* __restrict__ v2b,
                  const float* __restrict__ s2w, const float* __restrict__ s2b,
                  const float* __restrict__ ln2g, const float* __restrict__ ln2b,
                  const float* __restrict__ clsw, const float* __restrict__ clsb,
                  float* __restrict__ out) {
    __shared__ float sx[NNODE];
    __shared__ float sq1[NNODE * 4], sk1[NNODE * 4], sv1[NNODE * 4], sy1[NNODE * 4];
    __shared__ float sq2[NNODE], sk2[NNODE], sv2[NNODE], sskip[NNODE], sy2[NNODE];
    __shared__ int   srow[NNODE + 1];
    __shared__ int   scol[MAXE];
    __shared__ float red[128];

    const int tid = threadIdx.x;
    const int g   = blockIdx.x;

    if (tid < NNODE) sx[tid] = x[(size_t)g * NNODE + tid];
    if (tid < NNODE + 1) srow[tid] = row_ptr[tid];
    const bool lcol = (E <= MAXE);
    if (lcol) for (int e = tid; e < E; e += 128) scol[e] = colg[e];
    __syncthreads();
    const int* __restrict__ ecol = lcol ? scol : colg;

    // ---- layer 1: q/k/v per (node, head): in_dim = 1 so these are scalars ----
    for (int idx = tid; idx < NNODE * 4; idx += 128) {
        int n = idx >> 2, h = idx & 3;
        float xv = sx[n];
        sq1[idx] = xv * q1w[h] + q1b[h];
        sk1[idx] = xv * k1w[h] + k1b[h];
        sv1[idx] = xv * v1w[h] + v1b[h];
    }
    __syncthreads();

    // ---- layer 1 attention: per-(target,head) two-pass softmax (C=1 => scale 1) ----
    for (int idx = tid; idx < NNODE * 4; idx += 128) {
        int n = idx >> 2, h = idx & 3;
        int e0 = srow[n], e1 = srow[n + 1];
        float q = sq1[idx];
        float m = -3.402823e38f;
        for (int e = e0; e < e1; ++e)
            m = fmaxf(m, q * sk1[(ecol[e] << 2) + h]);
        float den = 0.f, acc = 0.f;
        for (int e = e0; e < e1; ++e) {
            int s = ecol[e];
            float w = expf(q * sk1[(s << 2) + h] - m);
            den += w;
            acc += w * sv1[(s << 2) + h];
        }
        float o = acc / den + sx[n] * s1w[h] + s1b[h];   // + root/skip
        sy1[idx] = LREL(o);
    }
    __syncthreads();

    // ---- layer 2: q/k/v/skip (in_dim = 4, H = 1, C = 1) ----
    if (tid < NNODE) {
        float a = sy1[tid * 4 + 0], b = sy1[tid * 4 + 1];
        float c = sy1[tid * 4 + 2], d = sy1[tid * 4 + 3];
        sq2[tid]   = a * q2w[0] + b * q2w[1] + c * q2w[2] + d * q2w[3] + q2b[0];
        sk2[tid]   = a * k2w[0] + b * k2w[1] + c * k2w[2] + d * k2w[3] + k2b[0];
        sv2[tid]   = a * v2w[0] + b * v2w[1] + c * v2w[2] + d * v2w[3] + v2b[0];
        sskip[tid] = a * s2w[0] + b * s2w[1] + c * s2w[2] + d * s2w[3] + s2b[0];
    }
    __syncthreads();

    if (tid < NNODE) {
        int e0 = srow[tid], e1 = srow[tid + 1];
        float q = sq2[tid];
        float m = -3.402823e38f;
        for (int e = e0; e < e1; ++e) m = fmaxf(m, q * sk2[ecol[e]]);
        float den = 0.f, acc = 0.f;
        for (int e = e0; e < e1; ++e) {
            int s = ecol[e];
            float w = expf(q * sk2[s] - m);
            den += w;
            acc += w * sv2[s];
        }
        float o = acc / den + sskip[tid];
        sy2[tid] = LREL(o);
    }
    __syncthreads();

    // ---- LayerNorm over 78 nodes + classifier (78 -> 2) ----
    red[tid] = (tid < NNODE) ? sy2[tid] : 0.f;
    __syncthreads();
    for (int s = 64; s > 0; s >>= 1) {
        if (tid < s) red[tid] += red[tid + s];
        __syncthreads();
    }
    float mean = red[0] * (1.0f / (float)NNODE);
    __syncthreads();

    float dd = (tid < NNODE) ? (sy2[tid] - mean) : 0.f;
    red[tid] = dd * dd;
    __syncthreads();
    for (int s = 64; s > 0; s >>= 1) {
        if (tid < s) red[tid] += red[tid + s];
        __syncthreads();
    }
    float rstd = rsqrtf(red[0] * (1.0f / (float)NNODE) + 1e-5f);
    __syncthreads();

    float yn = (tid < NNODE) ? ((sy2[tid] - mean) * rstd * ln2g[tid] + ln2b[tid]) : 0.f;
    for (int c = 0; c < 2; ++c) {
        red[tid] = (tid < NNODE) ? yn * clsw[c * NNODE + tid] : 0.f;
        __syncthreads();
        for (int s = 64; s > 0; s >>= 1) {
            if (tid < s) red[tid] += red[tid + s];
            __syncthreads();
        }
        if (tid == 0) out[(size_t)g * 2 + c] = red[0] + clsb[c];
        __syncthreads();
    }
}

// ---------------------------------------------------------------------------
extern "C" void kernel_launch(void* const* d_in, const int* in_sizes, int n_in,
                              void* d_out, int out_size, void* d_ws, size_t ws_size,
                              hipStream_t stream) {
    const float* cli   = (const float*)d_in[0];
    const float* radio = (const float*)d_in[1];
    const int*   ei    = (const int*)d_in[2];
    const float* ln1g  = (const float*)d_in[3];
    const float* ln1b  = (const float*)d_in[4];
    const float* encw  = (const float*)d_in[5];
    const float* encb  = (const float*)d_in[6];
    const float* q1w   = (const float*)d_in[7];
    const float* q1b   = (const float*)d_in[8];
    const float* k1w   = (const float*)d_in[9];
    const float* k1b   = (const float*)d_in[10];
    const float* v1w   = (const float*)d_in[11];
    const float* v1b   = (const float*)d_in[12];
    const float* s1w   = (const float*)d_in[13];
    const float* s1b   = (const float*)d_in[14];
    const float* q2w   = (const float*)d_in[15];
    const float* q2b   = (const float*)d_in[16];
    const float* k2w   = (const float*)d_in[17];
    const float* k2b   = (const float*)d_in[18];
    const float* v2w   = (const float*)d_in[19];
    const float* v2b   = (const float*)d_in[20];
    const float* s2w   = (const float*)d_in[21];
    const float* s2b   = (const float*)d_in[22];
    const float* ln2g  = (const float*)d_in[23];
    const float* ln2b  = (const float*)d_in[24];
    const float* clsw  = (const float*)d_in[25];
    const float* clsb  = (const float*)d_in[26];

    const int Bn = in_sizes[0] / 58;   // 8192
    const int E  = in_sizes[2] / 2;    // edge count

    // Workspace carve-up: x features [Bn x 78] f32, then CSR (row_ptr + cols).
    char*  ws      = (char*)d_ws;
    float* xfeat   = (float*)ws;
    int*   row_ptr = (int*)(ws + (size_t)Bn * NNODE * sizeof(float));
    int*   colbuf  = row_ptr + 128;

    build_csr_kernel<<<1, 64, 0, stream>>>(ei, E, row_ptr, colbuf);

    encode_kernel<<<Bn / 16, 64, 0, stream>>>(radio, cli, ln1g, ln1b, encw, encb, xfeat);

    graph_kernel<<<Bn, 128, 0, stream>>>(xfeat, row_ptr, colbuf, E,
                                         q1w, q1b, k1w, k1b, v1w, v1b, s1w, s1b,
                                         q2w, q2b, k2w, k2b, v2w, v2b, s2w, s2b,
                                         ln2g, ln2b, clsw, clsb, (float*)d_out);
}